// SparseDictAE_78365973283478
// MI455X (gfx1250) — compile-verified
//
#include <hip/hip_runtime.h>
#include <math.h>

// ---- problem constants (match reference) ----
#define NUM_EMB 256
#define EMB_DIM 64
#define KSEL    4
#define NBINS   33
#define COEF_MAX 2.0f
#define COMMIT   0.25f
#define EPSN     1e-10f
#define BZ 32
#define HH 32
#define WW 32
#define NSIG (BZ*HH*WW)      // 32768 signals
#define HWN  (HH*WW)         // 1024
#define STILE 128            // signals per workgroup

typedef __attribute__((ext_vector_type(2))) float v2f;
typedef __attribute__((ext_vector_type(8))) float v8f;

// -------------------------------------------------------------------
// Kernel 1: normalize dictionary columns -> DnT[atom][chan] (transposed,
// so WMMA A operands are contiguous b64 loads); zero the loss slot.
// -------------------------------------------------------------------
__global__ void k_norm(const float* __restrict__ dict,
                       float* __restrict__ DnT,
                       float* __restrict__ loss_out) {
    int a = blockIdx.x * blockDim.x + threadIdx.x;
    if (a == 0) *loss_out = 0.0f;   // runs before k_omp atomics (same stream)
    if (a < NUM_EMB) {
        float s = 0.0f;
        for (int c = 0; c < EMB_DIM; ++c) {
            float v = dict[c * NUM_EMB + a];
            s += v * v;
        }
        float inv = 1.0f / fmaxf(sqrtf(s), EPSN);
        for (int c = 0; c < EMB_DIM; ++c)
            DnT[a * EMB_DIM + c] = dict[c * NUM_EMB + a] * inv;
    }
}

// -------------------------------------------------------------------
// Kernel 2: Gram matrix G = Dn^T Dn (256x256); rows of DnT are contiguous.
// -------------------------------------------------------------------
__global__ void k_gram(const float* __restrict__ DnT, float* __restrict__ G) {
    int i = blockIdx.x;     // 0..255
    int j = threadIdx.x;    // 0..255
    float s = 0.0f;
    for (int c = 0; c < EMB_DIM; ++c)
        s += DnT[i * EMB_DIM + c] * DnT[j * EMB_DIM + c];
    G[i * NUM_EMB + j] = s;
}

// -------------------------------------------------------------------
// Kernel 3: fused correlation (WMMA f32 16x16x4) + per-lane OMP +
//           quantize + reconstruct + loss reduction
// Grid: NSIG/STILE blocks of 256 threads; dyn LDS = hbar[256][128] + red[8]
// -------------------------------------------------------------------
__global__ __launch_bounds__(256, 1)
void k_omp(const float* __restrict__ z_e,
           const float* __restrict__ DnT,
           const float* __restrict__ G,
           float* __restrict__ zq_out,
           float* __restrict__ loss_out,
           int*   __restrict__ tok_out) {
    extern __shared__ float lds[];
    float* hbar = lds;                       // [NUM_EMB][STILE]
    float* red  = lds + NUM_EMB * STILE;     // [8] wave partial sums

    const int tid  = threadIdx.x;
    const int lane = tid & 31;
    const int wave = tid >> 5;               // 0..7 (wave32)
    const int sigbase = blockIdx.x * STILE;

    // ============ Phase A: hbar[a][s] = sum_c Dn[c][a] * X[c][sigbase+s]
    // Each wave owns a 16-signal strip; V_WMMA_F32_16X16X4_F32, K-loop of 16.
    {
        const int n0    = sigbase + wave * 16;
        const int lhalf = lane >> 4;         // 0: lanes 0-15, 1: lanes 16-31
        const int l15   = lane & 15;
        const int nsig  = n0 + l15;
        const int nb    = nsig >> 10;        // batch index
        const int nhw   = nsig & (HWN - 1);  // h*W+w
        const float* xcol = z_e + (size_t)nb * EMB_DIM * HWN + nhw;

        // Hoist B operands (depend only on k-step): 16 x v2f = 32 VGPRs.
        // B (4x16 X tile): VGPR0 = K k0 (lanes 0-15) / k0+2 (lanes 16-31);
        // VGPR1 = k0+1 / k0+3 (ISA 32-bit operand layout).
        v2f Bops[16];
#pragma unroll
        for (int kk = 0; kk < 16; ++kk) {
            const int c0 = 4 * kk + 2 * lhalf;
            Bops[kk].x = xcol[(size_t)c0 * HWN];
            Bops[kk].y = xcol[(size_t)(c0 + 1) * HWN];
        }

        // Two atom tiles in flight -> two independent WMMA accumulator chains.
        for (int at = 0; at < NUM_EMB; at += 32) {
            v8f acc0 = {0.f,0.f,0.f,0.f,0.f,0.f,0.f,0.f};
            v8f acc1 = {0.f,0.f,0.f,0.f,0.f,0.f,0.f,0.f};
            const int atom0 = at + l15;
            const int atom1 = at + 16 + l15;
#pragma unroll
            for (int kk = 0; kk < 16; ++kk) {
                const int c0 = 4 * kk + 2 * lhalf;
                // A (16x4 Dn^T tile): contiguous pair in DnT -> one b64 load.
                v2f A0 = *(const v2f*)(DnT + (size_t)atom0 * EMB_DIM + c0);
                v2f A1 = *(const v2f*)(DnT + (size_t)atom1 * EMB_DIM + c0);
                acc0 = __builtin_amdgcn_wmma_f32_16x16x4_f32(
                        false, A0, false, Bops[kk], (short)0, acc0, false, false);
                acc1 = __builtin_amdgcn_wmma_f32_16x16x4_f32(
                        false, A1, false, Bops[kk], (short)0, acc1, false, false);
            }
            // C layout: lane(0-15)=N, VGPR r -> M=r (lanes 0-15) / M=r+8
            const int scol = wave * 16 + l15;
#pragma unroll
            for (int r = 0; r < 8; ++r) {
                hbar[(at + r + 8 * lhalf) * STILE + scol]      = acc0[r];
                hbar[(at + 16 + r + 8 * lhalf) * STILE + scol] = acc1[r];
            }
        }
    }
    __syncthreads();

    // ============ Phase B: one signal per lane (threads 0..127)
    float laneErr = 0.0f;
    if (tid < STILE) {
        const int s = tid;
        const int n = sigbase + s;
        const int nb  = n >> 10;
        const int nhw = n & (HWN - 1);

        int   I[KSEL];
        float x[KSEL];
        float hsel[KSEL];
        float Lm[10];            // packed lower-tri 4x4: row r starts r*(r+1)/2

        for (int k = 0; k < KSEL; ++k) {
            // argmax_a | hbar[a] - sum_{j<k} x_j * G[I_j][a] | (skip selected)
            int best = 0; float bestv = -1.0f;
            for (int a = 0; a < NUM_EMB; ++a) {
                float v = hbar[a * STILE + s];
                for (int j = 0; j < k; ++j)
                    v -= x[j] * G[I[j] * NUM_EMB + a];
                bool used = false;
                for (int j = 0; j < k; ++j) used |= (I[j] == a);
                float av = used ? -1.0f : fabsf(v);
                if (av > bestv) { bestv = av; best = a; }
            }
            I[k] = best;
            hsel[k] = hbar[best * STILE + s];
            // warm the G row this lane streams during the next argmax sweep
            __builtin_prefetch(&G[best * NUM_EMB], 0, 1);

            // incremental Cholesky of G[I,I] (diag == 1, Dn normalized)
            if (k == 0) {
                Lm[0] = 1.0f;
            } else {
                float w[3]; float ss = 0.0f;
                for (int r = 0; r < k; ++r) {
                    float v = G[I[r] * NUM_EMB + best];
                    for (int c = 0; c < r; ++c) v -= Lm[r*(r+1)/2 + c] * w[c];
                    v /= Lm[r*(r+1)/2 + r];
                    w[r] = v; ss += v * v;
                }
                float wc = sqrtf(fmaxf(1.0f - ss, 1e-12f));
                for (int c = 0; c < k; ++c) Lm[k*(k+1)/2 + c] = w[c];
                Lm[k*(k+1)/2 + k] = wc;
            }
            // forward solve L y = hsel, back solve L^T x = y (size k+1)
            float y[KSEL];
            for (int r = 0; r <= k; ++r) {
                float v = hsel[r];
                for (int c = 0; c < r; ++c) v -= Lm[r*(r+1)/2 + c] * y[c];
                y[r] = v / Lm[r*(r+1)/2 + r];
            }
            for (int r = k; r >= 0; --r) {
                float v = y[r];
                for (int c = r + 1; c <= k; ++c) v -= Lm[c*(c+1)/2 + r] * x[c];
                x[r] = v / Lm[r*(r+1)/2 + r];
            }
        }

        // quantize coefficients, emit tokens
        float cq[KSEL];
        for (int k = 0; k < KSEL; ++k) {
            float c = fminf(fmaxf(x[k], -COEF_MAX), COEF_MAX);
            float binf = (c + COEF_MAX) * ((NBINS - 1) / (2.0f * COEF_MAX));
            int bi = (int)rintf(binf);                  // RNE == jnp.round
            bi = bi < 0 ? 0 : (bi > NBINS - 1 ? NBINS - 1 : bi);
            cq[k] = -COEF_MAX + bi * (2.0f * COEF_MAX / (NBINS - 1));
            tok_out[(size_t)n * KSEL + k] = I[k] * NBINS + bi;
        }

        // reconstruct z_q from contiguous DnT rows, accumulate squared error
        for (int c = 0; c < EMB_DIM; ++c) {
            float zq = 0.0f;
            for (int k = 0; k < KSEL; ++k)
                zq += cq[k] * DnT[(size_t)I[k] * EMB_DIM + c];
            const size_t off = ((size_t)nb * EMB_DIM + c) * HWN + nhw;
            const float d = zq - z_e[off];
            laneErr += d * d;
            zq_out[off] = zq;
        }
    }

    // ============ block loss reduction -> one atomicAdd per block
    float v = laneErr;
    for (int off = 16; off > 0; off >>= 1)
        v += __shfl_down(v, off, 32);
    if (lane == 0) red[wave] = v;
    __syncthreads();
    if (tid == 0) {
        float t = 0.0f;
        for (int wv = 0; wv < 8; ++wv) t += red[wv];
        // loss = (1 + COMMIT) * SSE / (NSIG * EMB_DIM)
        atomicAdd(loss_out, t * ((1.0f + COMMIT) / (float)((size_t)NSIG * EMB_DIM)));
    }
}

// -------------------------------------------------------------------
extern "C" void kernel_launch(void* const* d_in, const int* in_sizes, int n_in,
                              void* d_out, int out_size, void* d_ws, size_t ws_size,
                              hipStream_t stream) {
    const float* z_e  = (const float*)d_in[0];   // [32,64,32,32]
    const float* dict = (const float*)d_in[1];   // [64,256]

    // workspace layout: DnT (256*64 f32) | G (256*256 f32)
    float* DnT = (float*)d_ws;
    float* G   = DnT + (size_t)NUM_EMB * EMB_DIM;

    // d_out layout: z_q_ste (2097152 f32) | loss (1 f32) | tokens (131072 i32)
    float* zq_out   = (float*)d_out;
    float* loss_out = zq_out + (size_t)NSIG * EMB_DIM;
    int*   tok_out  = (int*)(loss_out + 1);

    k_norm<<<1, 256, 0, stream>>>(dict, DnT, loss_out);
    k_gram<<<NUM_EMB, NUM_EMB, 0, stream>>>(DnT, G);

    const size_t shmem = (size_t)(NUM_EMB * STILE + 8) * sizeof(float); // 128KB + red
    k_omp<<<NSIG / STILE, 256, shmem, stream>>>(z_e, DnT, G, zq_out, loss_out, tok_out);
}